// MeanMaxPooling_84473416778475
// MI455X (gfx1250) — compile-verified
//
#include <hip/hip_runtime.h>

typedef __attribute__((ext_vector_type(2))) float v2f;
typedef __attribute__((ext_vector_type(4))) float f4;
typedef __attribute__((ext_vector_type(8))) float v8f;
typedef __attribute__((ext_vector_type(4))) int   i4;

#define N_      4
#define E_      64
#define L_      512
#define D_      256
#define KC      128          // K chunk staged in LDS
#define CHUNKS  (L_ / KC)    // 4
#define ETILE   16           // entities per workgroup
#define DTILE   64           // d-columns per workgroup (16 per wave)
#define SDS     (DTILE + 4)  // padded LDS row stride (68 floats = 17*16B)
#define THREADS 128          // 4 waves (wave32)

#define GAS __attribute__((address_space(1)))
#define LAS __attribute__((address_space(3)))

#if defined(__gfx1250__) && \
    __has_builtin(__builtin_amdgcn_global_load_async_to_lds_b128) && \
    __has_builtin(__builtin_amdgcn_s_wait_asynccnt)
#define HAVE_ASYNC 1
#else
#define HAVE_ASYNC 0
#endif

// Stage one 16-byte packet global -> LDS. Async path bypasses VGPRs and is
// tracked with ASYNCcnt; fallback is a plain load/store pair.
__device__ __forceinline__ void stage16(const float* __restrict__ g, float* l) {
#if HAVE_ASYNC
    __builtin_amdgcn_global_load_async_to_lds_b128((GAS i4*)g, (LAS i4*)l, 0, 0);
#else
    *(f4*)l = *(const f4*)g;
#endif
}

// Wait until at most N async packets (per wave: 16 per staged chunk) remain.
__device__ __forceinline__ void wait_async_16() {
#if HAVE_ASYNC
    __builtin_amdgcn_s_wait_asynccnt(16);
#endif
}
__device__ __forceinline__ void wait_async_0() {
#if HAVE_ASYNC
    __builtin_amdgcn_s_wait_asynccnt(0);
#endif
}

__global__ __launch_bounds__(THREADS)
void meanmax_pool_wmma(const float* __restrict__ doc,    // N x L x D
                       const float* __restrict__ mask,   // N x E x L
                       const float* __restrict__ nlen,   // N x E
                       float* __restrict__ out)          // N x E x 2D
{
    extern __shared__ float lds[];
    float* sMask = lds;                    // [ETILE][L_]           32 KB
    float* sDocB = lds + ETILE * L_;       // [2][KC][SDS] dbl-buf  ~70 KB

    const int tid  = threadIdx.x;
    const int lane = tid & 31;
    const int wave = tid >> 5;

    const int bx    = blockIdx.x;          // 0..63
    const int n     = bx >> 4;
    const int et    = (bx >> 2) & 3;
    const int dt    = bx & 3;
    const int ebase = et * ETILE;
    const int dbase = dt * DTILE;

    // ---- prologue: stage mask tile (16 async b128 per wave) + doc chunk 0 ----
    for (int i = tid; i < ETILE * (L_ / 4); i += THREADS) {
        const int el = i >> 7;             // / (L_/4)
        const int kq = (i & 127) * 4;
        stage16(&mask[(((size_t)n * E_ + ebase + el) * L_) + kq],
                &sMask[el * L_ + kq]);
    }
    for (int i = tid; i < KC * (DTILE / 4); i += THREADS) {
        const int r  = i >> 4;
        const int c4 = (i & 15) * 4;
        stage16(&doc[(((size_t)n * L_ + r) * D_) + dbase + c4],
                &sDocB[r * SDS + c4]);
    }

    const int h    = lane >> 4;            // half-wave select
    const int mn   = lane & 15;            // A-row (e) / C-col (d)
    const int dcol = wave * 16 + mn;       // this wave's d column in the 64-wide tile

    v8f csum = {0.f, 0.f, 0.f, 0.f, 0.f, 0.f, 0.f, 0.f};
    float cmax[8];
    #pragma unroll
    for (int r = 0; r < 8; ++r) cmax[r] = -__builtin_inff();

    for (int kc = 0; kc < CHUNKS; ++kc) {
        // All waves are done reading buffer (kc+1)&1 before we refill it.
        __syncthreads();

        if (kc + 1 < CHUNKS) {
            // Issue next chunk into the other buffer (16 async b128 per wave)...
            float* dst = sDocB + ((kc + 1) & 1) * (KC * SDS);
            for (int i = tid; i < KC * (DTILE / 4); i += THREADS) {
                const int r  = i >> 4;
                const int c4 = (i & 15) * 4;
                stage16(&doc[(((size_t)n * L_ + (kc + 1) * KC + r) * D_) + dbase + c4],
                        &dst[r * SDS + c4]);
            }
            // ...then drain everything EXCEPT those 16: async ops complete in
            // order, so chunk kc (and the mask tile) are now resident.
            wait_async_16();
        } else {
            wait_async_0();
        }
        __syncthreads();                   // chunk kc visible to all waves

        const float* sDoc = sDocB + (kc & 1) * (KC * SDS);

        // ---- sum-pool as GEMM: v_wmma_f32_16x16x4_f32, K swept in steps of 4 ----
        const float* aRow = &sMask[mn * L_ + kc * KC];
        #pragma unroll 4
        for (int kk = 0; kk < KC; kk += 4) {
            // A 16x4 f32: lane row = mn, K = kk + 2*h + {0,1} (contiguous pair)
            v2f a = *(const v2f*)&aRow[kk + 2 * h];
            // B 4x16 f32: lane col = dcol, same K pair (strided LDS reads)
            v2f b;
            b.x = sDoc[(kk + 2 * h + 0) * SDS + dcol];
            b.y = sDoc[(kk + 2 * h + 1) * SDS + dcol];
            csum = __builtin_amdgcn_wmma_f32_16x16x4_f32(
                false, a, false, b, (short)0, csum, false, false);
        }

        // ---- masked max-pool over the same chunk (C-fragment-shaped accs) ----
        #pragma unroll 2
        for (int kk = 0; kk < KC; kk += 4) {
            const float v0 = sDoc[(kk + 0) * SDS + dcol];
            const float v1 = sDoc[(kk + 1) * SDS + dcol];
            const float v2 = sDoc[(kk + 2) * SDS + dcol];
            const float v3 = sDoc[(kk + 3) * SDS + dcol];
            #pragma unroll
            for (int r = 0; r < 8; ++r) {
                const int m = r + 8 * h;   // e-row, matches WMMA C layout
                const f4 mk = *(const f4*)&sMask[m * L_ + kc * KC + kk];
                const float t0 = fmaxf(mk.x * v0, mk.y * v1);
                const float t1 = fmaxf(mk.z * v2, mk.w * v3);
                cmax[r] = fmaxf(cmax[r], fmaxf(t0, t1));
            }
        }
    }

    // ---- epilogue: out[n][e][0:256] = max, out[n][e][256:512] = sum/safe_len ----
    #pragma unroll
    for (int r = 0; r < 8; ++r) {
        const int m = r + 8 * h;
        const int e = ebase + m;
        const int d = dbase + wave * 16 + mn;
        const float len = nlen[n * E_ + e];
        const float sl  = (len > 0.f) ? len : 1.f;
        const size_t o  = ((size_t)(n * E_ + e)) * (2 * D_);
        out[o + d]      = cmax[r];
        out[o + D_ + d] = csum[r] / sl;
    }
}

extern "C" void kernel_launch(void* const* d_in, const int* in_sizes, int n_in,
                              void* d_out, int out_size, void* d_ws, size_t ws_size,
                              hipStream_t stream) {
    const float* doc  = (const float*)d_in[0];   // doc_state     N*L*D
    const float* mask = (const float*)d_in[1];   // nodes_mapping N*E*L
    const float* nlen = (const float*)d_in[2];   // nodes_len     N*E
    float* out = (float*)d_out;                  // N*E*2D

    const dim3 grid(N_ * (E_ / ETILE) * (D_ / DTILE));  // 64 workgroups
    const dim3 block(THREADS);
    const size_t shmem = (size_t)(ETILE * L_ + 2 * KC * SDS) * sizeof(float); // 100 KB

    hipLaunchKernelGGL(meanmax_pool_wmma, grid, block, shmem, stream,
                       doc, mask, nlen, out);
}